// NonMaximumSuppression_69561290326678
// MI455X (gfx1250) — compile-verified
//
#include <hip/hip_runtime.h>
#include <stdint.h>

#define BB 8
#define NN 50000
#define CCW 84        // detection row width (4 box + 80 cls)
#define NCLS 80
#define TOPK 1000
#define MAXB 300
#define TILE_ROWS 128
#define NMS_T 0.4f
#define NEG_INF_F (-1000000000.0f)

typedef unsigned int uint;
typedef unsigned long long u64;
typedef uint v4u __attribute__((ext_vector_type(4)));
typedef uint v8u __attribute__((ext_vector_type(8)));

__device__ __forceinline__ void tdm_wait0() {
#if __has_builtin(__builtin_amdgcn_s_wait_tensorcnt)
  __builtin_amdgcn_s_wait_tensorcnt(0);
#else
  asm volatile("s_wait_tensorcnt 0x0" ::: "memory");
#endif
}

__device__ __forceinline__ uint flipf(float f) {
  uint u = __float_as_uint(f);
  return (u & 0x80000000u) ? ~u : (u | 0x80000000u);
}
__device__ __forceinline__ float unflipf(uint m) {
  uint u = (m & 0x80000000u) ? (m & 0x7FFFFFFFu) : ~m;
  return __uint_as_float(u);
}
__device__ __forceinline__ u64 shflx64(u64 v, int m) {
  uint lo = (uint)v, hi = (uint)(v >> 32);
  lo = __shfl_xor(lo, m, 32);
  hi = __shfl_xor(hi, m, 32);
  return (((u64)hi) << 32) | (u64)lo;
}

// ---------------------------------------------------------------------------
// Kernel 1: per-box score = max over 80 class columns.
// TDM streams a 128x80 f32 tile (row stride 84 elems) from HBM into LDS.
// ---------------------------------------------------------------------------
__global__ __launch_bounds__(256) void score_kernel(const float* __restrict__ det,
                                                    float* __restrict__ scores) {
  // Single shared array in this kernel -> LDS offset 0 for the TDM descriptor.
  __shared__ __align__(16) float tile[TILE_ROWS * NCLS];  // 40 KB

  const int tilesPerImg = (NN + TILE_ROWS - 1) / TILE_ROWS;
  const int b = blockIdx.x / tilesPerImg;
  const int tb = blockIdx.x % tilesPerImg;
  const int r0 = tb * TILE_ROWS;
  const uint rows = (uint)((NN - r0) < TILE_ROWS ? (NN - r0) : TILE_ROWS);

  if (threadIdx.x < 32) {  // wave 0 issues one TDM op for the workgroup
    unsigned long long ga =
        (unsigned long long)(const void*)(det + ((size_t)b * NN + (size_t)r0) * CCW + 4);
    v4u g0, gz;
    v8u g1;
    gz[0] = 0u; gz[1] = 0u; gz[2] = 0u; gz[3] = 0u;
    // Group 0: count=1, lds_addr=0, global_addr, type=2
    g0[0] = 1u;
    g0[1] = 0u;
    g0[2] = (uint)(ga & 0xFFFFFFFFull);
    g0[3] = ((uint)((ga >> 32) & 0x1FFFFFFull)) | (2u << 30);
    // Group 1: data_size=4B, 2D tile 80 x rows, tensor row stride 84
    g1[0] = 0x00020000u;                   // data_size=2 (4 bytes)
    g1[1] = ((uint)NCLS) << 16;            // tensor_dim0 = 80 (bits 79:48 lo)
    g1[2] = (rows & 0xFFFFu) << 16;        // tensor_dim1 lo (bits 111:80 lo)
    g1[3] = ((uint)NCLS) << 16;            // tile_dim0 = 80 (bits 127:112)
    g1[4] = rows & 0xFFFFu;                // tile_dim1 (bits 143:128), tile_dim2=0
    g1[5] = (uint)CCW;                     // tensor_dim0_stride = 84 (bits 207:160)
    g1[6] = 0u;
    g1[7] = 0u;
    asm volatile("tensor_load_to_lds %0, %1, %2, %3"
                 :
                 : "s"(g0), "s"(g1), "s"(gz), "s"(gz)
                 : "memory");
    tdm_wait0();
  }
  __syncthreads();

  const int r = threadIdx.x >> 1;     // 2 threads per row
  const int half = threadIdx.x & 1;   // 40 classes each
  const float4* p = (const float4*)&tile[r * NCLS + half * 40];
  float m = -1.0f;
#pragma unroll
  for (int j = 0; j < 10; ++j) {
    float4 v = p[j];
    m = fmaxf(m, fmaxf(fmaxf(v.x, v.y), fmaxf(v.z, v.w)));
  }
  m = fmaxf(m, __shfl_xor(m, 1, 32));
  if (half == 0 && (r0 + r) < NN) scores[(size_t)b * NN + r0 + r] = m;
}

// ---------------------------------------------------------------------------
// Kernel 2 (one 1024-thread block per image): exact top-1000 (jax top_k
// semantics), bitonic sort to reference order, 300-step soft-NMS, outputs.
// ---------------------------------------------------------------------------
__global__ __launch_bounds__(1024) void nms_kernel(const float* __restrict__ det,
                                                   const float* __restrict__ scores,
                                                   float* __restrict__ out) {
  const int b = blockIdx.x;
  const int t = threadIdx.x;
  const int lane = t & 31;
  const int wid = t >> 5;
  const float* sc = scores + (size_t)b * NN;

  __shared__ uint hist[8 * 256];
  __shared__ u64 sKey[1024];
  __shared__ uint cand[1024];
  __shared__ float bx1[1024], by1[1024], bx2[1024], by2[1024], bar[1024];
  __shared__ uint tIdx[1024];
  __shared__ float tScore[1024];
  __shared__ uint selIdx[MAXB];
  __shared__ float selScore[MAXB];
  __shared__ u64 warpRed[32];
  __shared__ u64 sWin;
  __shared__ uint sSel, sKk, gcnt, ccnt, sMin;

  // ---- Phase A: radix-select the 1000th score (exact float bits) ----
  uint prefix = 0u, pmask = 0u, k = (uint)TOPK;
  for (int pass = 0; pass < 4; ++pass) {
    const int shift = 24 - 8 * pass;
    for (int i = t; i < 8 * 256; i += 1024) hist[i] = 0u;
    __syncthreads();
    uint* myh = &hist[(wid & 7) * 256];
    for (int n = t; n < NN; n += 1024) {
      uint u = __float_as_uint(sc[n]);
      if ((u & pmask) == prefix) atomicAdd(&myh[(u >> shift) & 255u], 1u);
    }
    __syncthreads();
    if (t < 256) {  // merge per-wave histograms (column t owned by thread t)
      uint s = 0u;
      for (int w = 0; w < 8; ++w) s += hist[w * 256 + t];
      hist[t] = s;
    }
    __syncthreads();
    if (t == 0) {
      uint c = 0u;
      for (int bin = 255; bin >= 0; --bin) {
        uint h = hist[bin];
        if (c + h >= k) { sSel = (uint)bin; sKk = k - c; break; }
        c += h;
      }
    }
    __syncthreads();
    prefix |= sSel << shift;
    pmask |= 0xFFu << shift;
    k = sKk;
  }
  const uint T = prefix;   // float bits of the 1000th-largest score
  const uint need = k;     // how many ==T to take (smallest indices first)

  // ---- Phase B: compaction into 64-bit keys (scorebits<<32)|~idx ----
  if (t == 0) { gcnt = 0u; ccnt = 0u; }
  __syncthreads();
  for (int n = t; n < NN; n += 1024) {
    uint u = __float_as_uint(sc[n]);
    if (u > T) {
      uint p = atomicAdd(&gcnt, 1u);
      sKey[p] = (((u64)u) << 32) | (u64)(uint)(~(uint)n);
    } else if (u == T) {
      uint p = atomicAdd(&ccnt, 1u);
      if (p < 1024u) cand[p] = (uint)n;
    }
  }
  __syncthreads();
  const uint base = gcnt;  // == TOPK - need
  const uint cN = ccnt < 1024u ? ccnt : 1024u;
  for (uint j = 0; j < need; ++j) {
    if (t == 0) { sMin = 0xFFFFFFFFu; sKey[base + j] = 0ull; }
    __syncthreads();
    uint myc = ((uint)t < cN) ? cand[t] : 0xFFFFFFFFu;
    if (myc != 0xFFFFFFFFu) atomicMin(&sMin, myc);
    __syncthreads();
    if (myc != 0xFFFFFFFFu && myc == sMin) {
      cand[t] = 0xFFFFFFFFu;
      sKey[base + j] = (((u64)T) << 32) | (u64)(uint)(~myc);
    }
    __syncthreads();
  }
  if (t >= TOPK) sKey[t] = 0ull;
  __syncthreads();

  // ---- Phase C: bitonic sort (ascending; read reversed => jax order) ----
  for (int ksz = 2; ksz <= 1024; ksz <<= 1) {
    for (int j = ksz >> 1; j > 0; j >>= 1) {
      int ixj = t ^ j;
      if (ixj > t) {
        u64 a = sKey[t], c2 = sKey[ixj];
        bool up = ((t & ksz) == 0);
        if (up ? (a > c2) : (a < c2)) { sKey[t] = c2; sKey[ixj] = a; }
      }
      __syncthreads();
    }
  }

  // ---- Phase D: extract sorted top-K, gather boxes ----
  float s_t;
  bool removed;
  {
    float x1 = 0.f, y1 = 0.f, x2 = 0.f, y2 = 0.f, ar = 0.f, scv = 0.f;
    uint oid = 0u;
    if (t < TOPK) {
      u64 key = sKey[1023 - t];
      oid = ~((uint)key);
      scv = __uint_as_float((uint)(key >> 32));
      const float4 bv = *(const float4*)(det + ((size_t)b * NN + oid) * CCW);
      x1 = bv.x; y1 = bv.y; x2 = bv.z; y2 = bv.w;
      ar = fmaxf(x2 - x1, 0.f) * fmaxf(y2 - y1, 0.f);
    }
    bx1[t] = x1; by1[t] = y1; bx2[t] = x2; by2[t] = y2; bar[t] = ar;
    tIdx[t] = oid; tScore[t] = scv;
    s_t = (t < TOPK) ? scv : NEG_INF_F;
    removed = (t >= TOPK);
  }
  __syncthreads();

  // ---- Phase E: 300-step soft-NMS scan ----
  const float myx1 = bx1[t], myy1 = by1[t], myx2 = bx2[t], myy2 = by2[t];
  const float myar = bar[t];
  for (int it = 0; it < MAXB; ++it) {
    u64 key = (((u64)flipf(s_t)) << 32) | (u64)(uint)(1023 - t);
#pragma unroll
    for (int m = 16; m >= 1; m >>= 1) { u64 o = shflx64(key, m); key = (key > o) ? key : o; }
    if (lane == 0) warpRed[wid] = key;
    __syncthreads();
    if (wid == 0) {
      u64 kk = warpRed[lane];
#pragma unroll
      for (int m = 16; m >= 1; m >>= 1) { u64 o = shflx64(kk, m); kk = (kk > o) ? kk : o; }
      if (lane == 0) sWin = kk;
    }
    __syncthreads();
    const u64 win = sWin;
    const int i = 1023 - (int)((uint)win);
    const float selS = unflipf((uint)(win >> 32));
    const float wx1 = bx1[i], wy1 = by1[i], wx2 = bx2[i], wy2 = by2[i], war = bar[i];
    float ix1 = fmaxf(wx1, myx1), iy1 = fmaxf(wy1, myy1);
    float ix2 = fminf(wx2, myx2), iy2 = fminf(wy2, myy2);
    float inter = fmaxf(ix2 - ix1, 0.f) * fmaxf(iy2 - iy1, 0.f);
    float uni = war + myar - inter;
    float iou = (uni > 0.f) ? (inter / uni) : 0.f;
    float w = (iou <= NMS_T) ? expf((-1.0f * iou) * iou) : 0.f;
    if (t == i) removed = true;
    s_t = removed ? NEG_INF_F : (s_t * w);
    if (t == 0) { selIdx[it] = (uint)i; selScore[it] = selS; }
  }
  __syncthreads();

  // ---- Phase F: outputs, one wave per selected row ----
  for (int r = wid; r < MAXB; r += 32) {
    const int i = (int)selIdx[r];
    const uint n0 = tIdx[i];
    const float ratio = selScore[r] / tScore[i];
    const float* crow = det + ((size_t)b * NN + n0) * CCW + 4;
    const int c1 = lane, c2 = lane + 32, c3 = lane + 64;
    float v1 = crow[c1], v2 = crow[c2];
    float v3 = (c3 < NCLS) ? crow[c3] : 0.f;
    u64 kk = (((u64)__float_as_uint(v1)) << 32) | (u64)(uint)(NCLS - 1 - c1);
    u64 k2 = (((u64)__float_as_uint(v2)) << 32) | (u64)(uint)(NCLS - 1 - c2);
    kk = (kk > k2) ? kk : k2;
    if (c3 < NCLS) {
      u64 k3 = (((u64)__float_as_uint(v3)) << 32) | (u64)(uint)(NCLS - 1 - c3);
      kk = (kk > k3) ? kk : k3;
    }
#pragma unroll
    for (int m = 16; m >= 1; m >>= 1) { u64 o = shflx64(kk, m); kk = (kk > o) ? kk : o; }
    const float label = (float)(NCLS - 1 - (int)((uint)kk));
    float* orow = out + ((size_t)b * MAXB + r) * 85;
    if (lane == 0) {
      orow[0] = bx1[i]; orow[1] = by1[i]; orow[2] = bx2[i]; orow[3] = by2[i];
      orow[4] = label;
    }
    orow[5 + c1] = v1 * ratio;
    orow[5 + c2] = v2 * ratio;
    if (c3 < NCLS) orow[5 + c3] = v3 * ratio;
  }
}

extern "C" void kernel_launch(void* const* d_in, const int* in_sizes, int n_in,
                              void* d_out, int out_size, void* d_ws, size_t ws_size,
                              hipStream_t stream) {
  (void)in_sizes; (void)n_in; (void)out_size; (void)ws_size;
  const float* det = (const float*)d_in[0];
  float* out = (float*)d_out;
  float* scores = (float*)d_ws;  // B*N floats = 1.6 MB scratch

  const int tilesPerImg = (NN + TILE_ROWS - 1) / TILE_ROWS;  // 391
  score_kernel<<<dim3(BB * tilesPerImg), dim3(256), 0, stream>>>(det, scores);
  nms_kernel<<<dim3(BB), dim3(1024), 0, stream>>>(det, scores, out);
}